// InverseChebyshevApproximation_15633680958242
// MI455X (gfx1250) — compile-verified
//
#include <hip/hip_runtime.h>

typedef __attribute__((ext_vector_type(2))) float v2f;
typedef __attribute__((ext_vector_type(8))) float v8f;

#define DEG 16
#define TPW 4            // column-tiles per wave
#define WAVES_PER_BLOCK 8

// out[m, n] = sum_d a[m, d] * T_d(x[n]),  m in [0,32), d in [0,16)
// GEMM via V_WMMA_F32_16X16X4_F32: D(16x16) += A(16x4) x B(4x16), 4 chained k-steps.
// CN = compile-time N (0 -> use runtime Nrt).
template <int CN>
__global__ __launch_bounds__(256) void cheb_wmma_kernel(
    const float* __restrict__ x, const float* __restrict__ a,
    float* __restrict__ out, int Nrt)
{
    const unsigned N = (CN > 0) ? (unsigned)CN : (unsigned)Nrt;

    const int  lane = threadIdx.x & 31;
    const int  wave = threadIdx.x >> 5;
    const bool hi   = lane >= 16;      // lane half
    const int  l16  = lane & 15;

    // ---- A-matrix fragments (coeffs), two M-tiles (rows 0-15, 16-31) ----
    // 32-bit A 16x4 layout: lanes 0-15 hold {A[M=l16][4k+0], A[M=l16][4k+1]},
    //                       lanes16-31 hold {A[M=l16][4k+2], A[M=l16][4k+3]}.
    v2f A0[4], A1[4];
    {
        const int kofs = hi ? 2 : 0;
        const float* r0 = a + (0  + l16) * DEG;
        const float* r1 = a + (16 + l16) * DEG;
#pragma unroll
        for (int k = 0; k < 4; ++k) {
            A0[k] = *(const v2f*)(r0 + 4 * k + kofs);
            A1[k] = *(const v2f*)(r1 + 4 * k + kofs);
        }
    }

#pragma unroll
    for (int i = 0; i < TPW; ++i) {
        // Wave-interleaved tiles: the 8 waves of a block cover 128 contiguous
        // columns at each i, so same-row stores merge in L2.
        const int tile = blockIdx.x * (WAVES_PER_BLOCK * TPW)
                       + i * WAVES_PER_BLOCK + wave;
        const unsigned n0 = (unsigned)tile * 16u;
        if (n0 >= N) break;            // wave-uniform guard (EXEC stays all-1s)

        // Lanes l and l+16 handle the SAME column -> no cross-lane data
        // movement needed to build the B fragment.
        const float xv = x[n0 + l16];
        const float x2 = 2.0f * xv;

        // Shifted Chebyshev recurrence: lo lanes generate T_k, hi lanes T_{k+2}.
        // s[k] = T_{k + 2*hi}(x); all indices compile-time -> stays in VGPRs.
        const float t2 = __builtin_fmaf(x2, xv, -1.0f);   // T2
        const float t3 = __builtin_fmaf(x2, t2, -xv);     // T3
        float s[14];
        s[0] = hi ? t2 : 1.0f;
        s[1] = hi ? t3 : xv;
#pragma unroll
        for (int k = 2; k < 14; ++k)
            s[k] = __builtin_fmaf(x2, s[k - 1], -s[k - 2]);

        v8f c0 = {}; // rows 0-15
        v8f c1 = {}; // rows 16-31
#pragma unroll
        for (int k = 0; k < 4; ++k) {
            // 32-bit B 4x16 layout: VGPR0 = K=4k (lanes 0-15) / K=4k+2 (16-31),
            //                       VGPR1 = K=4k+1            / K=4k+3
            v2f B;
            B.x = s[4 * k + 0];
            B.y = s[4 * k + 1];
            c0 = __builtin_amdgcn_wmma_f32_16x16x4_f32(
                     false, A0[k], false, B, (short)0, c0, false, false);
            c1 = __builtin_amdgcn_wmma_f32_16x16x4_f32(
                     false, A1[k], false, B, (short)0, c1, false, false);
        }

        // C/D layout: VGPR j holds row (j + 8*hi) at column n0 + l16.
        // All offsets fit in 32 bits (out is 256 MB max here) -> SADDR stores.
        const unsigned col  = n0 + (unsigned)l16;
        const unsigned base = (hi ? 8u * N : 0u) + col;
#pragma unroll
        for (int j = 0; j < 8; ++j) {
            __builtin_nontemporal_store(c0[j], out + base + (unsigned)j * N);
            __builtin_nontemporal_store(c1[j], out + base + (unsigned)(j + 16) * N);
        }
    }
}

extern "C" void kernel_launch(void* const* d_in, const int* in_sizes, int n_in,
                              void* d_out, int out_size, void* d_ws, size_t ws_size,
                              hipStream_t stream) {
    const float* x = (const float*)d_in[0];   // [N] float32
    const float* a = (const float*)d_in[1];   // [32,16] float32
    float* out = (float*)d_out;               // [32,N] float32
    const int N = in_sizes[0];

    const int tiles = (N + 15) / 16;
    const int tiles_per_block = WAVES_PER_BLOCK * TPW; // 32
    const int blocks = (tiles + tiles_per_block - 1) / tiles_per_block;

    if (N == 2097152) {
        cheb_wmma_kernel<2097152>
            <<<blocks, WAVES_PER_BLOCK * 32, 0, stream>>>(x, a, out, N);
    } else {
        cheb_wmma_kernel<0>
            <<<blocks, WAVES_PER_BLOCK * 32, 0, stream>>>(x, a, out, N);
    }
}